// EdgeDecoder_86792699118121
// MI455X (gfx1250) — compile-verified
//
#include <hip/hip_runtime.h>

typedef float v2f __attribute__((ext_vector_type(2)));
typedef float v4f __attribute__((ext_vector_type(4)));
typedef float v8f __attribute__((ext_vector_type(8)));

#define D_DIM 128

// ---------------------------------------------------------------------------
// Phase 1: per-node projections via V_WMMA_F32_16X16X4_F32 (full f32).
//   gk[n] = h[n]·Wk + b    gi[n] = h[n]·Wi
// One wave handles a 16-node tile. A = h tile (16x4 per step), B column 0 = Wk
// chunk, column 1 = Wi chunk (zero elsewhere), accumulated over 32 K-chunks.
// A layout (ISA 7.12.2, 32-bit A 16x4): lanes 0-15 = M, VGPR0/1 = K0/K1;
// lanes 16-31 = M, VGPR0/1 = K2/K3  -> contiguous float2 load per lane.
// D layout: VGPR r, lanes 0-15 -> (M=r, N=lane); lanes 16-31 -> (M=8+r).
// B is built branchlessly: unconditional float2 load from a loop-invariant
// per-lane pointer + v_cndmask select, so the WMMA loop has no exec-mask
// round trips and loads pipeline freely.
// ---------------------------------------------------------------------------
__global__ __launch_bounds__(256)
void node_proj_wmma(const float* __restrict__ h,
                    const float* __restrict__ W,   // (1, 384): [Wk | Wi | We]
                    const float* __restrict__ b,
                    float* __restrict__ gk,
                    float* __restrict__ gi,
                    int n_nodes)
{
    const int lane = threadIdx.x & 31;
    const int tile = blockIdx.x * (blockDim.x >> 5) + (threadIdx.x >> 5);
    const int n0   = tile * 16;
    if (n0 >= n_nodes) return;              // wave-uniform: EXEC stays full

    const int col  = lane & 15;             // M for A rows, N for B columns
    const int half = lane >> 4;             // 0 -> K{0,1}, 1 -> K{2,3}

    int row = n0 + col;
    if (row >= n_nodes) row = n_nodes - 1;  // clamp loads, keep EXEC full
    const float* arow = h + (size_t)row * D_DIM + 2 * half;

    // Loop-invariant per-lane weight pointer: col 0 -> Wk, col 1 -> Wi,
    // cols >= 2 read Wk (even) or Wi (odd) harmlessly (selected away below). Always in-bounds.
    const float* wrow  = W + (col & 1) * D_DIM + 2 * half;
    const bool   live  = (col <= 1);

    v8f acc = {0.f, 0.f, 0.f, 0.f, 0.f, 0.f, 0.f, 0.f};
#pragma unroll
    for (int c = 0; c < D_DIM / 4; ++c) {
        v2f A  = *(const v2f*)(arow + 4 * c);   // global_load_b64 (h tile)
        v2f wv = *(const v2f*)(wrow + 4 * c);   // global_load_b64 (weights, cached)
        v2f B;
        B.x = live ? wv.x : 0.f;                // v_cndmask, no branch
        B.y = live ? wv.y : 0.f;
        acc = __builtin_amdgcn_wmma_f32_16x16x4_f32(
            /*neg_a=*/false, A, /*neg_b=*/false, B,
            /*c_mod=*/(short)0, acc, /*reuse_a=*/false, /*reuse_b=*/false);
    }

    const float bias = b[0];
    const int mbase = n0 + half * 8;        // lanes 0/1: M=0..7, lanes 16/17: M=8..15
    if (col == 0) {                         // D column 0 -> gk (+ bias folded in)
#pragma unroll
        for (int r = 0; r < 8; ++r)
            if (mbase + r < n_nodes) gk[mbase + r] = acc[r] + bias;
    } else if (col == 1) {                  // D column 1 -> gi
#pragma unroll
        for (int r = 0; r < 8; ++r)
            if (mbase + r < n_nodes) gi[mbase + r] = acc[r];
    }
}

// ---------------------------------------------------------------------------
// Phase 2: bandwidth-bound edge stream (256 MB of e read exactly once).
// One lane per edge. Non-temporal loads/stores for the e stream and output so
// the 192 MB L2 keeps gk/gi (and src/dst lines) resident for the gathers.
// We (uniform) is scalar-promoted by the compiler.
// ---------------------------------------------------------------------------
__global__ __launch_bounds__(256)
void edge_decode(const float* __restrict__ e,
                 const long long* __restrict__ src,
                 const long long* __restrict__ dst,
                 const float* __restrict__ W,
                 const float* __restrict__ gk,   // includes +b
                 const float* __restrict__ gi,
                 float* __restrict__ out,
                 int n_edges)
{
    const int idx = blockIdx.x * blockDim.x + threadIdx.x;
    if (idx >= n_edges) return;

    const float* We = W + 2 * D_DIM;                        // [256,384)
    const v4f*  er  = (const v4f*)(e + (size_t)idx * D_DIM);

    float acc = 0.f;
#pragma unroll
    for (int j = 0; j < D_DIM / 4; ++j) {
        v4f v = __builtin_nontemporal_load(er + j);         // streaming: NT
        const float w0 = We[4 * j + 0];
        const float w1 = We[4 * j + 1];
        const float w2 = We[4 * j + 2];
        const float w3 = We[4 * j + 3];
        acc = fmaf(v.x, w0, acc);
        acc = fmaf(v.y, w1, acc);
        acc = fmaf(v.z, w2, acc);
        acc = fmaf(v.w, w3, acc);
    }

    const int s = (int)src[idx];
    const int d = (int)dst[idx];
    const float logit = acc + gk[s] + gi[d];                // b already in gk
    const float y = 1.0f / (1.0f + __expf(-logit));
    __builtin_nontemporal_store(y, out + idx);
}

// ---------------------------------------------------------------------------
// Host-side launch. Input order: h, e, src, dst, W, b.
// ---------------------------------------------------------------------------
extern "C" void kernel_launch(void* const* d_in, const int* in_sizes, int n_in,
                              void* d_out, int out_size, void* d_ws, size_t ws_size,
                              hipStream_t stream)
{
    const float*     h   = (const float*)d_in[0];
    const float*     e   = (const float*)d_in[1];
    const long long* src = (const long long*)d_in[2];
    const long long* dst = (const long long*)d_in[3];
    const float*     W   = (const float*)d_in[4];
    const float*     b   = (const float*)d_in[5];

    const int n_nodes = in_sizes[0] / D_DIM;
    const int n_edges = in_sizes[2];

    float* gk = (float*)d_ws;              // n_nodes floats (dot(h,Wk)+b)
    float* gi = gk + n_nodes;              // n_nodes floats (dot(h,Wi))

    // Phase 1: 16 nodes per wave, 8 waves per 256-thread block.
    const int tiles   = (n_nodes + 15) / 16;
    const int blocks1 = (tiles + 7) / 8;
    node_proj_wmma<<<blocks1, 256, 0, stream>>>(h, W, b, gk, gi, n_nodes);

    // Phase 2: one lane per edge (stream ordering guarantees phase 1 is done).
    const int blocks2 = (n_edges + 255) / 256;
    edge_decode<<<blocks2, 256, 0, stream>>>(e, src, dst, W, gk, gi,
                                             (float*)d_out, n_edges);
}